// MANO_NEW_49735721287738
// MI455X (gfx1250) — compile-verified
//
#include <hip/hip_runtime.h>
#include <math.h>
#include <stdint.h>

typedef __attribute__((ext_vector_type(2))) float v2f;
typedef __attribute__((ext_vector_type(8))) float v8f;

#define B_N     8192
#define NV      778
#define NV3     2334      // NV*3
#define NV3P    2336      // padded to multiple of 32 (2 n-tiles per block)
#define KP      136       // 135 pose features padded to multiple of 4
#define LDSK    140       // LDS row stride: 16B-aligned rows for async B128, conflict-free
#define NJ      16
#define NVP     784       // 49*16 padded vertex count

// ---------------- workspace layout (floats) ----------------
#define OFF_VSHAPED  0                                  // NV3P
#define OFF_JK       (OFF_VSHAPED + NV3P)               // 48
#define OFF_WPAD     (OFF_JK + 48)                      // NVP*16
#define OFF_PT       (OFF_WPAD + NVP*16)                // NV3P*KP (transposed padded posedirs)
#define OFF_PF       (OFF_PT + NV3P*KP)                 // B_N*KP  (16B-aligned: dword off %4==0)
#define OFF_AOUT     (OFF_PF + (size_t)B_N*KP)          // B_N*256 (transposed padded A)
#define OFF_VPOSED   (OFF_AOUT + (size_t)B_N*256)       // B_N*NV3P
#define OFF_VERTS    (OFF_VPOSED + (size_t)B_N*NV3P)    // B_N*NV3

// ---------------- CDNA5 async global->LDS copy (ASYNCcnt-tracked) ----------------
__device__ inline void async_b128_to_lds(uint32_t lds_byte_off, const float* gaddr) {
    asm volatile("global_load_async_to_lds_b128 %0, %1, off"
                 :: "v"(lds_byte_off), "v"(gaddr)
                 : "memory");
}
__device__ inline void wait_asynccnt0() {
    asm volatile("s_wait_asynccnt 0x0" ::: "memory");
}

// ---------------- helpers ----------------
__device__ inline void quat2R(float qw, float qx, float qy, float qz, float R[9]) {
    float inv = rsqrtf(qw*qw + qx*qx + qy*qy + qz*qz);
    qw *= inv; qx *= inv; qy *= inv; qz *= inv;
    float w2 = qw*qw, x2 = qx*qx, y2 = qy*qy, z2 = qz*qz;
    float wx = qw*qx, wy = qw*qy, wz = qw*qz;
    float xy = qx*qy, xz = qx*qz, yz = qy*qz;
    R[0] = w2 + x2 - y2 - z2; R[1] = 2.f*xy - 2.f*wz;     R[2] = 2.f*wy + 2.f*xz;
    R[3] = 2.f*wz + 2.f*xy;   R[4] = w2 - x2 + y2 - z2;   R[5] = 2.f*yz - 2.f*wx;
    R[6] = 2.f*xz - 2.f*wy;   R[7] = 2.f*wx + 2.f*yz;     R[8] = w2 - x2 - y2 + z2;
}

__device__ inline void rodrigues(const float* t3, float R[9]) {
    float tx = t3[0], ty = t3[1], tz = t3[2];
    float ax = tx + 1e-8f, ay = ty + 1e-8f, az = tz + 1e-8f;
    float angle = sqrtf(ax*ax + ay*ay + az*az);
    float inv   = 1.0f / angle;
    float half  = 0.5f * angle;
    float sh = __sinf(half), ch = __cosf(half);
    quat2R(ch, sh*tx*inv, sh*ty*inv, sh*tz*inv, R);
}

// ---------------- K0a: v_shaped (batch invariant, zero-padded) ----------------
__global__ void k0a_vshaped(const float* __restrict__ user_shape,
                            const float* __restrict__ shapedirs,
                            const float* __restrict__ v_template,
                            float* __restrict__ ws) {
    int n = blockIdx.x * blockDim.x + threadIdx.x;
    if (n >= NV3P) return;
    float acc = 0.f;
    if (n < NV3) {
        #pragma unroll
        for (int s = 0; s < 10; ++s) acc += user_shape[s] * shapedirs[s*NV3 + n];
        acc += v_template[n];
    }
    ws[OFF_VSHAPED + n] = acc;
}

// ---------------- K0b: Jk[16][3] (batch invariant) ----------------
__global__ void k0b_joints(const float* __restrict__ Jreg, float* __restrict__ ws) {
    int t = threadIdx.x;
    if (t >= 48) return;
    int j = t / 3, c = t % 3;
    const float* vs = ws + OFF_VSHAPED;
    float acc = 0.f;
    for (int v = 0; v < NV; ++v) acc += vs[v*3 + c] * Jreg[v*21 + j];
    ws[OFF_JK + t] = acc;
}

// ---------------- K0c: transpose+pad posedirs -> Pt[NV3P][KP] ----------------
__global__ void k0c_padposedirs(const float* __restrict__ posedirs,
                                float* __restrict__ ws) {
    int idx = blockIdx.x * blockDim.x + threadIdx.x;
    if (idx >= NV3P * KP) return;
    int n = idx / KP, k = idx % KP;
    float v = (n < NV3 && k < 135) ? posedirs[(size_t)k * NV3 + n] : 0.f;
    ws[OFF_PT + idx] = v;
}

// ---------------- K0d: pad weights -> Wp[NVP][16] ----------------
__global__ void k0d_padweights(const float* __restrict__ weights,
                               float* __restrict__ ws) {
    int idx = blockIdx.x * blockDim.x + threadIdx.x;
    if (idx >= NVP * NJ) return;
    int v = idx / NJ;
    ws[OFF_WPAD + idx] = (v < NV) ? weights[idx] : 0.f;
}

// ---------------- K1: rodrigues + pose feature + FK chain ----------------
// writes A transposed/padded: At[b][n=k*4+l (16 rows, 12 used)][j (16)]
__global__ void k1_pose_chain(const float* __restrict__ theta,
                              const float* __restrict__ dq,
                              const float* __restrict__ iq,
                              float* __restrict__ ws) {
    int b = blockIdx.x * blockDim.x + threadIdx.x;
    if (b >= B_N) return;
    const float* th = theta + b*45;
    float* pf = ws + OFF_PF + (size_t)b*KP;

    for (int j = 0; j < 15; ++j) {
        float R[9]; rodrigues(th + 3*j, R);
        #pragma unroll
        for (int e = 0; e < 9; ++e)
            pf[j*9 + e] = R[e] - ((e % 4 == 0) ? 1.f : 0.f);
    }
    pf[135] = 0.f;  // K padding

    float Jk[48];
    #pragma unroll
    for (int i = 0; i < 48; ++i) Jk[i] = ws[OFF_JK + i];

    float* Ab = ws + OFF_AOUT + (size_t)b*256;
    // zero pad rows n = 12..15
    #pragma unroll
    for (int i = 192; i < 256; ++i) Ab[i] = 0.f;

    // root rotation: qmul(delta_quat, init_quat) -> R
    float q1w = dq[b*4+0], q1x = dq[b*4+1], q1y = dq[b*4+2], q1z = dq[b*4+3];
    float q2w = iq[0], q2x = iq[1], q2y = iq[2], q2z = iq[3];
    float qw = q1w*q2w - q1x*q2x - q1y*q2y - q1z*q2z;
    float qx = q1w*q2x + q1x*q2w + q1y*q2z - q1z*q2y;
    float qy = q1w*q2y - q1x*q2z + q1y*q2w + q1z*q2x;
    float qz = q1w*q2z + q1x*q2y - q1y*q2x + q1z*q2w;
    float G0r[9]; quat2R(qw, qx, qy, qz, G0r);
    float G0t[3] = { Jk[0], Jk[1], Jk[2] };

    #pragma unroll
    for (int k = 0; k < 3; ++k) {
        Ab[(k*4+0)*16 + 0] = G0r[k*3+0];
        Ab[(k*4+1)*16 + 0] = G0r[k*3+1];
        Ab[(k*4+2)*16 + 0] = G0r[k*3+2];
        Ab[(k*4+3)*16 + 0] = G0t[k] -
            (G0r[k*3+0]*Jk[0] + G0r[k*3+1]*Jk[1] + G0r[k*3+2]*Jk[2]);
    }

    // 5 chains of 3 joints: {1,2,3},{4,5,6},{7,8,9},{10,11,12},{13,14,15}
    for (int c = 0; c < 5; ++c) {
        float Gr[9], Gt[3];
        #pragma unroll
        for (int i = 0; i < 9; ++i) Gr[i] = G0r[i];
        Gt[0] = G0t[0]; Gt[1] = G0t[1]; Gt[2] = G0t[2];
        int j = c*3 + 1;
        for (int s = 0; s < 3; ++s, ++j) {
            float R[9]; rodrigues(th + 3*(j-1), R);
            int p = (s == 0) ? 0 : (j - 1);
            float rel[3] = { Jk[j*3+0] - Jk[p*3+0],
                             Jk[j*3+1] - Jk[p*3+1],
                             Jk[j*3+2] - Jk[p*3+2] };
            float nR[9], nT[3];
            #pragma unroll
            for (int k = 0; k < 3; ++k) {
                #pragma unroll
                for (int l = 0; l < 3; ++l)
                    nR[k*3+l] = Gr[k*3+0]*R[0*3+l] + Gr[k*3+1]*R[1*3+l] + Gr[k*3+2]*R[2*3+l];
                nT[k] = Gr[k*3+0]*rel[0] + Gr[k*3+1]*rel[1] + Gr[k*3+2]*rel[2] + Gt[k];
            }
            #pragma unroll
            for (int i = 0; i < 9; ++i) Gr[i] = nR[i];
            Gt[0] = nT[0]; Gt[1] = nT[1]; Gt[2] = nT[2];
            #pragma unroll
            for (int k = 0; k < 3; ++k) {
                Ab[(k*4+0)*16 + j] = Gr[k*3+0];
                Ab[(k*4+1)*16 + j] = Gr[k*3+1];
                Ab[(k*4+2)*16 + j] = Gr[k*3+2];
                Ab[(k*4+3)*16 + j] = Gt[k] -
                    (Gr[k*3+0]*Jk[j*3+0] + Gr[k*3+1]*Jk[j*3+1] + Gr[k*3+2]*Jk[j*3+2]);
            }
        }
    }
}

// ---------------- K2: WMMA GEMM1  v_posed = Pf @ posedirs + v_shaped ----------------
// block = 2 waves; 16x136 Pf tile staged via ASYNC global->LDS B128 DMA
// (row stride 140 dwords: 16B-aligned, bank-conflict-free); all fragment
// loads unconditional b64; K fully unrolled.
__global__ void k2_gemm_pose(float* __restrict__ ws) {
    __shared__ float ldsA[16 * LDSK];
    const float* Pf = ws + OFF_PF;
    const float* Pt = ws + OFF_PT;
    const float* vs = ws + OFF_VSHAPED;
    float* vposed   = ws + OFF_VPOSED;

    int tid  = threadIdx.x;
    int b0   = blockIdx.y * 16;

    // async stage of the 16x136 A tile as 544 B128 chunks (34 per row)
    const float* pfbase = Pf + (size_t)b0 * KP;
    for (int i = tid; i < 16 * (KP/4); i += 64) {
        int r = i / (KP/4), c4 = i % (KP/4);
        uint32_t ldst = (uint32_t)(uintptr_t)&ldsA[r * LDSK + c4 * 4];
        async_b128_to_lds(ldst, pfbase + r * KP + c4 * 4);
    }
    wait_asynccnt0();
    __syncthreads();

    int wave = tid >> 5;
    int lane = tid & 31;
    int half = lane >> 4;        // 0: K+{0,1}   1: K+{2,3}
    int lm   = lane & 15;
    int n    = blockIdx.x * 32 + wave * 16 + lm;   // < NV3P always

    const float* ldrow = &ldsA[lm * LDSK];
    const float* ptrow = Pt + (size_t)n * KP;

    v8f acc = {};
    #pragma unroll
    for (int k0 = 0; k0 < KP; k0 += 4) {
        int ka = k0 + half*2;
        v2f a  = *(const v2f*)(ldrow + ka);   // 8B aligned: stride 140, ka even
        v2f bm = *(const v2f*)(ptrow + ka);   // 8B aligned: stride 136, ka even
        acc = __builtin_amdgcn_wmma_f32_16x16x4_f32(false, a, false, bm,
                                                    (short)0, acc, false, false);
    }

    float add = vs[n];
    #pragma unroll
    for (int i = 0; i < 8; ++i) {
        int b = b0 + half*8 + i;              // C layout: vgpr i -> M = i (+8 hi lanes)
        vposed[(size_t)b * NV3P + n] = acc[i] + add;
    }
}

// ---------------- K3: WMMA GEMM2 (skinning) + vertex apply ----------------
// one wave = (batch b, 16-vertex tile); unconditional b64 fragment loads.
__global__ void k3_skin(float* __restrict__ ws) {
    __shared__ float S[256];
    const float* Wp   = ws + OFF_WPAD;
    const float* Aout = ws + OFF_AOUT;
    const float* vposed = ws + OFF_VPOSED;
    float* verts = ws + OFF_VERTS;

    int lane = threadIdx.x;
    int half = lane >> 4;
    int lm   = lane & 15;
    int v0   = blockIdx.x * 16;
    int b    = blockIdx.y;
    const float* Ab = Aout + (size_t)b * 256;   // At[n][j], rows 12..15 zero

    const float* wrow = Wp + (v0 + lm) * NJ;    // padded: always in range
    const float* arow = Ab + lm * 16;

    v8f acc = {};
    #pragma unroll
    for (int k0 = 0; k0 < NJ; k0 += 4) {
        int ka = k0 + half*2;                   // joint index (K dim), even
        v2f a  = *(const v2f*)(wrow + ka);
        v2f bm = *(const v2f*)(arow + ka);
        acc = __builtin_amdgcn_wmma_f32_16x16x4_f32(false, a, false, bm,
                                                    (short)0, acc, false, false);
    }

    // stage C tile in LDS: row = vertex-in-tile, col = (k*4+l)
    #pragma unroll
    for (int i = 0; i < 8; ++i)
        S[(i + half*8)*16 + lm] = acc[i];
    __syncthreads();

    if (lane < 16) {
        int vv = v0 + lane;
        if (vv < NV) {
            const float* vp = vposed + (size_t)b * NV3P + 3*vv;
            float x = vp[0], y = vp[1], z = vp[2];
            float* out = verts + (size_t)b * NV3 + 3*vv;
            #pragma unroll
            for (int k = 0; k < 3; ++k) {
                const float* Sr = &S[lane*16 + k*4];
                out[k] = Sr[0]*x + Sr[1]*y + Sr[2]*z + Sr[3];
            }
        }
    }
}

// ---------------- K4: joints = J_regressor^T @ verts + trans ----------------
__global__ void k4_joints(const float* __restrict__ Jreg,
                          const float* __restrict__ init_trans,
                          const float* __restrict__ ws,
                          float* __restrict__ out) {
    __shared__ float lv[NV3];
    int b = blockIdx.x;
    const float* verts = ws + OFF_VERTS + (size_t)b * NV3;
    for (int i = threadIdx.x; i < NV3; i += blockDim.x) lv[i] = verts[i];
    __syncthreads();
    int t = threadIdx.x;
    if (t < 63) {
        int j = t / 3, c = t % 3;
        float acc = 0.f;
        for (int v = 0; v < NV; ++v) acc += lv[v*3 + c] * Jreg[v*21 + j];
        out[(size_t)b*63 + t] = acc + init_trans[b*3 + c];
    }
}

// ---------------- launcher ----------------
extern "C" void kernel_launch(void* const* d_in, const int* in_sizes, int n_in,
                              void* d_out, int out_size, void* d_ws, size_t ws_size,
                              hipStream_t stream) {
    const float* theta      = (const float*)d_in[0];
    const float* delta_quat = (const float*)d_in[1];
    const float* user_shape = (const float*)d_in[2];
    const float* init_quat  = (const float*)d_in[3];
    const float* init_trans = (const float*)d_in[4];
    const float* v_template = (const float*)d_in[5];
    const float* shapedirs  = (const float*)d_in[6];
    const float* J_regressor= (const float*)d_in[7];
    const float* posedirs   = (const float*)d_in[8];
    const float* weights    = (const float*)d_in[9];
    float* ws  = (float*)d_ws;
    float* out = (float*)d_out;

    k0a_vshaped<<<(NV3P + 255)/256, 256, 0, stream>>>(user_shape, shapedirs, v_template, ws);
    k0b_joints <<<1, 64, 0, stream>>>(J_regressor, ws);
    k0c_padposedirs<<<(NV3P*KP + 255)/256, 256, 0, stream>>>(posedirs, ws);
    k0d_padweights <<<(NVP*NJ + 255)/256, 256, 0, stream>>>(weights, ws);
    k1_pose_chain<<<B_N/64, 64, 0, stream>>>(theta, delta_quat, init_quat, ws);
    k2_gemm_pose<<<dim3(NV3P/32, B_N/16), 64, 0, stream>>>(ws);
    k3_skin<<<dim3(NVP/16, B_N), 32, 0, stream>>>(ws);
    k4_joints<<<B_N, 96, 0, stream>>>(J_regressor, init_trans, ws, out);
}